// NeuralSplineTransformer_11063835754615
// MI455X (gfx1250) — compile-verified
//
#include <hip/hip_runtime.h>
#include <stdint.h>

// Problem constants (fixed by the reference)
#define B_DIM   4096
#define F_DIM   512
#define NBIN    16
#define NPAR    47          // 3*NBIN - 1
#define HALF    256         // features handled per block
#define BDIM    256         // threads per block (8 wave32)
#define TILE_DW (NPAR * HALF)     // 12032 dwords staged in LDS
#define TILE_XF (TILE_DW / 4)     // 3008 b128 transfers

__global__ void __launch_bounds__(BDIM)
spline_kernel(const float* __restrict__ x,
              const float* __restrict__ params,
              const float* __restrict__ x0,
              const float* __restrict__ xf,
              float* __restrict__ out_y,
              float* __restrict__ ws_partial)
{
    __shared__ float tile[TILE_DW];   // 48,128 B
    __shared__ float red[BDIM];       //  1,024 B

    const int tid   = threadIdx.x;
    const int half  = blockIdx.x;     // 0 or 1
    const int b     = blockIdx.y;     // 0..4095
    const int fbase = half * HALF;

    // ---------------- async stage params[b, 0:47, fbase:fbase+256] -> LDS ----
    const float* gbase = params + (size_t)b * NPAR * F_DIM;
    const uint32_t lds_base = (uint32_t)(uintptr_t)(&tile[0]); // flat low 32 = LDS addr

#pragma unroll
    for (int i = 0; i < 12; ++i) {
        int g = i * BDIM + tid;              // b128 transfer index
        if (g < TILE_XF) {
            int row = g >> 6;                // 64 x b128 per 256-float row
            int c16 = g & 63;
            uint32_t voff  = (uint32_t)(row * (F_DIM * 4) + fbase * 4 + c16 * 16);
            uint32_t laddr = lds_base + (uint32_t)(g * 16);
            asm volatile("global_load_async_to_lds_b128 %0, %1, %2"
                         :: "v"(laddr), "v"(voff), "s"(gbase)
                         : "memory");
        }
    }
    asm volatile("s_wait_asynccnt 0" ::: "memory");
    __syncthreads();

    // ---------------- per-element spline ----------------
    const int f = fbase + tid;
    const float x0v = x0[f];
    const float xfv = xf[f];
    const float rng = xfv - x0v;            // == yf - y0
    const float xv  = x[(size_t)b * F_DIM + f];

    float w[NBIN], h[NBIN], sp[NBIN + 1];

    // widths = softmax(p[:, 0:16]) * rng
    float m = -1e30f;
#pragma unroll
    for (int j = 0; j < NBIN; ++j) { w[j] = tile[j * HALF + tid]; m = fmaxf(m, w[j]); }
    float s = 0.f;
#pragma unroll
    for (int j = 0; j < NBIN; ++j) { w[j] = expf(w[j] - m); s += w[j]; }
    float inv = rng / s;
#pragma unroll
    for (int j = 0; j < NBIN; ++j) w[j] *= inv;

    // heights = softmax(p[:, 16:32]) * rng
    m = -1e30f;
#pragma unroll
    for (int j = 0; j < NBIN; ++j) { h[j] = tile[(NBIN + j) * HALF + tid]; m = fmaxf(m, h[j]); }
    s = 0.f;
#pragma unroll
    for (int j = 0; j < NBIN; ++j) { h[j] = expf(h[j] - m); s += h[j]; }
    inv = rng / s;
#pragma unroll
    for (int j = 0; j < NBIN; ++j) h[j] *= inv;

    // slopes = [1, softplus(p[:, 32:47]), 1]
    sp[0] = 1.f;
    sp[NBIN] = 1.f;
#pragma unroll
    for (int j = 0; j < NBIN - 1; ++j) {
        float v = tile[(2 * NBIN + j) * HALF + tid];
        sp[j + 1] = (v > 20.f) ? v : log1pf(expf(v));
    }

    // cumsum knots + fused bin search / gather (selects via cndmask, no scratch)
    float cx = x0v, cy = x0v;
    float wsel = w[0], hsel = h[0], kx = x0v, ky = x0v, d0 = sp[0], d1 = sp[1];
#pragma unroll
    for (int j = 1; j < NBIN; ++j) {
        cx += w[j - 1];
        cy += h[j - 1];
        bool p = xv > cx;                    // bin index >= j
        wsel = p ? w[j]      : wsel;
        hsel = p ? h[j]      : hsel;
        kx   = p ? cx        : kx;
        ky   = p ? cy        : ky;
        d0   = p ? sp[j]     : d0;
        d1   = p ? sp[j + 1] : d1;
    }

    const float sl   = hsel / wsel;
    const float eps  = (xv - kx) / wsel;
    const float omge = 1.f - eps;
    const float e1e  = eps * omge;
    const float e2   = eps * eps;
    const float den  = sl + (d1 + d0 - 2.f * sl) * e1e;
    const float yv   = ky + hsel * (sl * e2 + d0 * e1e) / den;
    const float dy   = sl * sl * (d1 * e2 + 2.f * sl * e1e + d0 * omge * omge)
                       / (den * den);

    __builtin_nontemporal_store(yv, &out_y[(size_t)b * F_DIM + f]);

    // ---------------- deterministic block reduction of log(dy/dx) -----------
    red[tid] = logf(dy);
    __syncthreads();
#pragma unroll
    for (int st = BDIM / 2; st > 0; st >>= 1) {
        if (tid < st) red[tid] += red[tid + st];
        __syncthreads();
    }
    if (tid == 0) ws_partial[b * 2 + half] = red[0];
}

__global__ void __launch_bounds__(256)
spline_finalize(const float* __restrict__ ws_partial, float* __restrict__ out_ld)
{
    int b = blockIdx.x * 256 + threadIdx.x;
    if (b < B_DIM) out_ld[b] = ws_partial[2 * b] + ws_partial[2 * b + 1];
}

extern "C" void kernel_launch(void* const* d_in, const int* in_sizes, int n_in,
                              void* d_out, int out_size, void* d_ws, size_t ws_size,
                              hipStream_t stream)
{
    const float* x      = (const float*)d_in[0];
    const float* params = (const float*)d_in[1];
    const float* x0     = (const float*)d_in[2];
    const float* xf     = (const float*)d_in[3];

    float* out_y  = (float*)d_out;                       // B*F floats
    float* out_ld = (float*)d_out + (size_t)B_DIM * F_DIM; // B floats
    float* wsp    = (float*)d_ws;                        // 2*B floats of scratch

    dim3 grid(2, B_DIM, 1);
    spline_kernel<<<grid, BDIM, 0, stream>>>(x, params, x0, xf, out_y, wsp);
    spline_finalize<<<(B_DIM + 255) / 256, 256, 0, stream>>>(wsp, out_ld);
}